// ConvCaps_77335181132239
// MI455X (gfx1250) — compile-verified
//
#include <hip/hip_runtime.h>

typedef float v2f __attribute__((ext_vector_type(2)));
typedef float v8f __attribute__((ext_vector_type(8)));

#define C_OUT   32
#define PSIZE   16
#define BK      288
#define N_TOTAL 576
#define NG      4        // patches per workgroup
#define WAVES   8
#define THREADS 256
#define EPS_    1e-8f
#define SQRT2_  1.41421356237309515f
#define LN_2PI_ 1.8378770664093453f
#define LAM0_   0.01f
#define LAM1_   1.0f
#define W_FLOATS (BK*C_OUT*PSIZE)     // 147456 weight floats

// ---- LDS layout (float offsets) ----
#define XV_OFF   0
#define XV_SZ    (NG*BK*PSIZE)        // 18432 : pose patches
#define AIN_OFF  (XV_OFF + XV_SZ)
#define AIN_SZ   (NG*BK)              // 1152  : input activations
#define RDS_OFF  (AIN_OFF + AIN_SZ)
#define RDS_SZ   (NG*BK*C_OUT)        // 36864 : routing R
#define SS_OFF   (RDS_OFF + RDS_SZ)
#define SS_SZ    (NG*C_OUT)           // 128   : sum R*a
#define MM_OFF   (SS_OFF + SS_SZ)
#define MM_SZ    (NG*C_OUT*PSIZE)     // 2048  : sum Ra*v -> mu
#define QQ_OFF   (MM_OFF + MM_SZ)
#define QQ_SZ    (NG*C_OUT*PSIZE)     // 2048  : sum Ra*v^2 -> 1/(sqrt2*ss)
#define CO_OFF   (QQ_OFF + QQ_SZ)
#define CO_SZ    (NG*C_OUT)           // 128   : cost accum
#define BS_OFF   (CO_OFF + CO_SZ)
#define BS_SZ    (NG*C_OUT*PSIZE)     // 2048  : 0.5*ln(2*pi*ss)
#define AO_OFF   (BS_OFF + BS_SZ)
#define AO_SZ    (NG*C_OUT)           // a_out
#define LA_OFF   (AO_OFF + AO_SZ)
#define LA_SZ    (NG*C_OUT)           // log(a_out)
#define LN_OFF   (LA_OFF + LA_SZ)
#define LN_SZ    (WAVES*NG*C_OUT)     // 1024  : per-wave ln_ap scratch
#define LDS_FLOATS (LN_OFF + LN_SZ)   // 64128 floats = 256512 B (< 320 KB/WGP)

// CDNA5 async copy: global memory -> LDS, 16B per lane, tracked by ASYNCcnt.
// VDST carries the LDS byte address (flat-address low 32 bits == LDS offset),
// VADDR carries the 64-bit global address.
__device__ __forceinline__ void async_load16_to_lds(const float* gsrc, float* ldst) {
    unsigned loff = (unsigned)(uintptr_t)ldst;
    unsigned long long ga = (unsigned long long)(uintptr_t)gsrc;
    asm volatile("global_load_async_to_lds_b128 %0, %1, off"
                 :: "v"(loff), "v"(ga) : "memory");
}
__device__ __forceinline__ void wait_asynccnt0() {
    asm volatile("s_wait_asynccnt 0x0" ::: "memory");
}

__global__ __launch_bounds__(THREADS, 1)
void ConvCapsEM_kernel(const float* __restrict__ x, const float* __restrict__ rr,
                       const float* __restrict__ W, const float* __restrict__ beta_a,
                       const float* __restrict__ beta_u, float* __restrict__ out)
{
    extern __shared__ float sm[];
    const int tid  = threadIdx.x;
    const int wv   = tid >> 5;
    const int lane = tid & 31;
    const int half = lane >> 4;       // which K-pair / which row block
    const int l15  = lane & 15;
    const int nA   = l15 >> 2;        // A-matrix: row M = 4*nA + iA
    const int iA   = l15 & 3;
    const int jb   = half * 2;        // K (=j) pair base
    const int cl   = l15 >> 2;        // D/B column: N = 4*cl + kk
    const int kk   = l15 & 3;
    const int n0   = blockIdx.x * NG;
    const float lam = LAM0_ + (LAM1_ - LAM0_) * rr[0];

    // Warm L2 with the (tiny, fully reused) weight tensor: global_prefetch_b8
    // covers one 128B line per call; stride threads across the whole tensor.
    for (int i = tid * 32; i < W_FLOATS; i += THREADS * 32)
        __builtin_prefetch(W + i, 0, 1);

    // ---- Phase A: gather the 4 patches (pose + activation) into LDS ----
    // Reference reshape maps: n = b*36 + s/9,  bk = 32*(s%9) + cap,
    // with s = (ki*3+kj)*36 + oh*6 + ow, pixel (h,w) = (2*oh+ki, 2*ow+kj)
    for (int seg = tid; seg < NG*BK; seg += THREADS) {
        int nl  = seg / BK;
        int bk  = seg - nl*BK;
        int n   = n0 + nl;
        int b   = n / 36;
        int q   = n - b*36;
        int r9  = bk >> 5;
        int cap = bk & 31;
        int s   = q*9 + r9;
        int kk2 = s / 36;
        int rem = s - kk2*36;
        int oh  = rem / 6;
        int ow  = rem - oh*6;
        int ki  = kk2 / 3;
        int kj  = kk2 - ki*3;
        long gbase = ((long)(b*14 + (2*oh + ki))*14 + (2*ow + kj)) * 544;
        const float* src = x + gbase + cap*PSIZE;
        float* dst = sm + XV_OFF + (nl*BK + bk)*PSIZE;
        // async copy 64B (pose 4x4) straight into LDS, no VGPR bounce
        async_load16_to_lds(src + 0,  dst + 0);
        async_load16_to_lds(src + 4,  dst + 4);
        async_load16_to_lds(src + 8,  dst + 8);
        async_load16_to_lds(src + 12, dst + 12);
        sm[AIN_OFF + nl*BK + bk] = x[gbase + 512 + cap];
    }
    for (int i = tid; i < RDS_SZ; i += THREADS) sm[RDS_OFF + i] = 1.0f / C_OUT;
    wait_asynccnt0();
    __syncthreads();

    for (int it = 0; it < 3; ++it) {
        // zero SS, MM, QQ, CO (contiguous)
        for (int i = tid; i < SS_SZ + MM_SZ + QQ_SZ + CO_SZ; i += THREADS)
            sm[SS_OFF + i] = 0.0f;
        __syncthreads();

        // ---- Pass 1: moments. Recompute v via WMMA f32 16x16x4. ----
        for (int cg = 0; cg < 8; ++cg) {
            const int c = cg*4 + cl;
            float accM[8], accQ[8];
            #pragma unroll
            for (int p = 0; p < 8; ++p) { accM[p] = 0.f; accQ[p] = 0.f; }
            float accS0 = 0.f, accS1 = 0.f;

            for (int bk = wv; bk < BK; bk += WAVES) {
                // A: xv[nA, bk, iA, jb..jb+1]  (contiguous pair in LDS)
                v2f a = *(const v2f*)(sm + XV_OFF + (nA*BK + bk)*PSIZE + iA*4 + jb);
                // B: W[bk, c, jb..jb+1, kk]   (stride 4 floats between j's)
                const float* wp = W + ((bk*C_OUT + c)*4 + jb)*4 + kk;
                v2f bb; bb.x = wp[0]; bb.y = wp[4];
                v8f d = {};
                d = __builtin_amdgcn_wmma_f32_16x16x4_f32(
                        false, a, false, bb, (short)0, d, false, false);

                const int nlo = 2*half, nhi = nlo + 1;
                float ra0 = sm[RDS_OFF + (nlo*BK + bk)*C_OUT + c] * sm[AIN_OFF + nlo*BK + bk];
                float ra1 = sm[RDS_OFF + (nhi*BK + bk)*C_OUT + c] * sm[AIN_OFF + nhi*BK + bk];
                if (cg == 0 && kk == 0) { accS0 += ra0; accS1 += ra1; }
                #pragma unroll
                for (int p = 0; p < 8; ++p) {
                    float ra = (p < 4) ? ra0 : ra1;
                    float vv = d[p];
                    accM[p] += ra * vv;
                    accQ[p] += ra * vv * vv;
                }
            }
            #pragma unroll
            for (int p = 0; p < 8; ++p) {
                int nl  = 2*half + (p >> 2);
                int ps  = (p & 3)*4 + kk;          // ps = i*4 + k
                int idx = (nl*C_OUT + c)*PSIZE + ps;
                atomicAdd(sm + MM_OFF + idx, accM[p]);
                atomicAdd(sm + QQ_OFF + idx, accQ[p]);
            }
            if (cg == 0 && kk == 0) {
                atomicAdd(sm + SS_OFF + (2*half)*C_OUT + c,     accS0);
                atomicAdd(sm + SS_OFF + (2*half + 1)*C_OUT + c, accS1);
            }
        }
        __syncthreads();

        // ---- Stats: mu, sigma^2, a_out ----
        for (int idx = tid; idx < MM_SZ; idx += THREADS) {
            int nc = idx >> 4;
            int c  = nc & 31;
            float s   = sm[SS_OFF + nc];
            float m   = sm[MM_OFF + idx];
            float qv  = sm[QQ_OFF + idx];
            float inv = 1.0f / (s + EPS_);
            float mu  = m * inv;
            float ss  = (qv - 2.0f*mu*m + mu*mu*s) * inv + EPS_;
            float lss = logf(ss);
            sm[MM_OFF + idx] = mu;
            sm[QQ_OFF + idx] = 1.0f / (SQRT2_ * ss);
            sm[BS_OFF + idx] = 0.5f * (LN_2PI_ + lss);
            atomicAdd(sm + CO_OFF + nc, beta_u[c] + 0.5f * lss);
        }
        __syncthreads();
        if (tid < NG*C_OUT) {
            int c = tid & 31;
            float s   = sm[SS_OFF + tid];
            float act = lam * (beta_a[c] - s * sm[CO_OFF + tid]);
            float ao  = 1.0f / (1.0f + expf(-act));
            sm[AO_OFF + tid] = ao;
            sm[LA_OFF + tid] = logf(ao);
        }
        __syncthreads();

        // ---- Pass 2: update R (skip on last iter) ----
        if (it < 2) {
            for (int bk = wv; bk < BK; bk += WAVES) {
                v2f a = *(const v2f*)(sm + XV_OFF + (nA*BK + bk)*PSIZE + iA*4 + jb);
                for (int cg = 0; cg < 8; ++cg) {
                    const int c = cg*4 + cl;
                    const float* wp = W + ((bk*C_OUT + c)*4 + jb)*4 + kk;
                    v2f bb; bb.x = wp[0]; bb.y = wp[4];
                    v8f d = {};
                    d = __builtin_amdgcn_wmma_f32_16x16x4_f32(
                            false, a, false, bb, (short)0, d, false, false);
                    float lp0 = 0.f, lp1 = 0.f;
                    #pragma unroll
                    for (int p = 0; p < 8; ++p) {
                        int nl  = 2*half + (p >> 2);
                        int ps  = (p & 3)*4 + kk;
                        int idx = (nl*C_OUT + c)*PSIZE + ps;
                        float diff = d[p] - sm[MM_OFF + idx];
                        float lp = -(diff*diff) * sm[QQ_OFF + idx] - sm[BS_OFF + idx];
                        if (p < 4) lp0 += lp; else lp1 += lp;
                    }
                    // sum over k (4 lanes sharing one c)
                    lp0 += __shfl_xor(lp0, 1, 32); lp0 += __shfl_xor(lp0, 2, 32);
                    lp1 += __shfl_xor(lp1, 1, 32); lp1 += __shfl_xor(lp1, 2, 32);
                    if (kk == 0) {
                        int nlo = 2*half;
                        sm[LN_OFF + wv*(NG*C_OUT) + nlo*C_OUT + c] =
                            lp0 + sm[LA_OFF + nlo*C_OUT + c];
                        sm[LN_OFF + wv*(NG*C_OUT) + (nlo + 1)*C_OUT + c] =
                            lp1 + sm[LA_OFF + (nlo + 1)*C_OUT + c];
                    }
                }
                __syncthreads();
                // softmax over the 32 output capsules: one capsule per lane
                #pragma unroll
                for (int nl2 = 0; nl2 < NG; ++nl2) {
                    float v = sm[LN_OFF + wv*(NG*C_OUT) + nl2*C_OUT + lane];
                    float mx = v;
                    mx = fmaxf(mx, __shfl_xor(mx, 16, 32));
                    mx = fmaxf(mx, __shfl_xor(mx,  8, 32));
                    mx = fmaxf(mx, __shfl_xor(mx,  4, 32));
                    mx = fmaxf(mx, __shfl_xor(mx,  2, 32));
                    mx = fmaxf(mx, __shfl_xor(mx,  1, 32));
                    float e = expf(v - mx);
                    float su = e;
                    su += __shfl_xor(su, 16, 32);
                    su += __shfl_xor(su,  8, 32);
                    su += __shfl_xor(su,  4, 32);
                    su += __shfl_xor(su,  2, 32);
                    su += __shfl_xor(su,  1, 32);
                    sm[RDS_OFF + (nl2*BK + bk)*C_OUT + lane] = e / su;
                }
                __syncthreads();
            }
        }
    }

    // ---- Output: concat(mu, a_out) per patch ----
    for (int idx = tid; idx < NG*544; idx += THREADS) {
        int nl = idx / 544;
        int ch = idx - nl*544;
        float val = (ch < 512) ? sm[MM_OFF + nl*512 + ch]
                               : sm[AO_OFF + nl*C_OUT + (ch - 512)];
        out[(long)(n0 + nl)*544 + ch] = val;
    }
}

extern "C" void kernel_launch(void* const* d_in, const int* in_sizes, int n_in,
                              void* d_out, int out_size, void* d_ws, size_t ws_size,
                              hipStream_t stream) {
    const float* x  = (const float*)d_in[0];
    const float* r  = (const float*)d_in[1];
    const float* w  = (const float*)d_in[2];
    const float* ba = (const float*)d_in[3];
    const float* bu = (const float*)d_in[4];
    float* out = (float*)d_out;
    (void)d_ws; (void)ws_size; (void)n_in; (void)in_sizes; (void)out_size;

    size_t lds = (size_t)LDS_FLOATS * sizeof(float);
    hipFuncSetAttribute((const void*)ConvCapsEM_kernel,
                        hipFuncAttributeMaxDynamicSharedMemorySize, (int)lds);
    ConvCapsEM_kernel<<<N_TOTAL / NG, THREADS, lds, stream>>>(x, r, w, ba, bu, out);
}